// Net_11536282157803
// MI455X (gfx1250) — compile-verified
//
#include <hip/hip_runtime.h>
#include <hip/hip_bf16.h>

typedef float v2f __attribute__((ext_vector_type(2)));
typedef float v4f __attribute__((ext_vector_type(4)));
typedef float v8f __attribute__((ext_vector_type(8)));

#define N_X     20000
#define N_GENE  4264
#define N_NODES (N_X + N_GENE)
#define NUM_REL 4

#define KC      32          // K-chunk staged in LDS
#define LDS_STR 34          // Bs row stride (floats): even (b64 align) + conflict-free

// ---------------------------------------------------------------------------
// WMMA f32 GEMM:  C[M,Nc] = A[M,K] @ B[K,Nc] (+bias) (optional relu)
// A is a virtual concat: rows [0,split) from A0, rows [split,M) from A1.
// Block = 256 threads = 8 waves. Block tile: 256 rows x 64 cols.
// Each wave: 32 rows x 64 cols via 2x4 accumulators of 16x16 f32
// (V_WMMA_F32_16X16X4_F32). B is staged transposed in LDS per 32-deep
// K-chunk so each B fragment is a single ds_load_b64.
// The K-remainder chunk is fully hoisted out of the main loop so the hot
// loop is single-path (no accumulator phi-copies at the loop join).
// ---------------------------------------------------------------------------
__global__ __launch_bounds__(256)
void wmma_gemm_f32(const float* __restrict__ A0, const float* __restrict__ A1,
                   int split,
                   const float* __restrict__ B, const float* __restrict__ bias,
                   float* __restrict__ C, int M, int K, int Nc, int do_relu)
{
    __shared__ float Bs[64][LDS_STR];   // Bs[n][k], transposed panel

    const int lane = threadIdx.x & 31;
    const int wave = threadIdx.x >> 5;
    const int m0   = blockIdx.y * 256 + wave * 32;
    const int n0   = blockIdx.x * 64;
    const int half = lane >> 4;         // 0: lanes 0-15, 1: lanes 16-31
    const int l16  = lane & 15;

    // Two A rows per lane (rows m0+l16 and m0+16+l16), clamped; masked at store
    int r0 = m0 + l16;        if (r0 >= M) r0 = M - 1;
    int r1 = m0 + 16 + l16;   if (r1 >= M) r1 = M - 1;
    const float* Arow0 = (r0 < split) ? (A0 + (size_t)r0 * K)
                                      : (A1 + (size_t)(r0 - split) * K);
    const float* Arow1 = (r1 < split) ? (A0 + (size_t)r1 * K)
                                      : (A1 + (size_t)(r1 - split) * K);

    // Cooperative B-panel fill indices: coalesced global rows
    const int nc = threadIdx.x & 63;     // column within panel
    const int kr = threadIdx.x >> 6;     // 0..3
    int colLd = n0 + nc; if (colLd >= Nc) colLd = Nc - 1;   // clamped (masked at store)
    const float* Bcol = B + colLd;       // column base: element k at Bcol[k*Nc]

    v8f acc[2][4];
#pragma unroll
    for (int i = 0; i < 2; ++i)
#pragma unroll
        for (int j = 0; j < 4; ++j) acc[i][j] = (v8f){};

    const int kfull = (K / KC) * KC;

    // =================== main loop: full chunks, guard-free ===================
    for (int k0 = 0; k0 < kfull; k0 += KC) {
        __syncthreads();
#pragma unroll
        for (int p = 0; p < 8; ++p) {
            const int kk = kr + p * 4;                    // 0..31, unique
            Bs[nc][kk] = Bcol[(size_t)(k0 + kk) * Nc];
        }
        __syncthreads();

#pragma unroll
        for (int s = 0; s < 8; ++s) {
            const int kk = s * 4 + (half << 1);
            const int kA = k0 + kk;
            v2f a0, a1;
            a0.x = Arow0[kA]; a0.y = Arow0[kA + 1];
            a1.x = Arow1[kA]; a1.y = Arow1[kA + 1];
#pragma unroll
            for (int j = 0; j < 4; ++j) {
                v2f b = *(const v2f*)&Bs[j * 16 + l16][kk];   // ds_load_b64
                acc[0][j] = __builtin_amdgcn_wmma_f32_16x16x4_f32(
                    false, a0, false, b, (short)0, acc[0][j], false, false);
                acc[1][j] = __builtin_amdgcn_wmma_f32_16x16x4_f32(
                    false, a1, false, b, (short)0, acc[1][j], false, false);
            }
        }
    }

    // =================== single tail chunk (K % KC != 0) ======================
    if (kfull < K) {
        __syncthreads();
#pragma unroll
        for (int p = 0; p < 8; ++p) {
            const int kk = kr + p * 4;
            const int k  = kfull + kk;
            Bs[nc][kk] = (k < K) ? Bcol[(size_t)k * Nc] : 0.0f;
        }
        __syncthreads();

#pragma unroll
        for (int s = 0; s < 8; ++s) {
            const int kk = s * 4 + (half << 1);
            const int kA = kfull + kk;
            v2f a0, a1;
            a0.x = (kA     < K) ? Arow0[kA]     : 0.0f;
            a0.y = (kA + 1 < K) ? Arow0[kA + 1] : 0.0f;
            a1.x = (kA     < K) ? Arow1[kA]     : 0.0f;
            a1.y = (kA + 1 < K) ? Arow1[kA + 1] : 0.0f;
#pragma unroll
            for (int j = 0; j < 4; ++j) {
                v2f b = *(const v2f*)&Bs[j * 16 + l16][kk];
                acc[0][j] = __builtin_amdgcn_wmma_f32_16x16x4_f32(
                    false, a0, false, b, (short)0, acc[0][j], false, false);
                acc[1][j] = __builtin_amdgcn_wmma_f32_16x16x4_f32(
                    false, a1, false, b, (short)0, acc[1][j], false, false);
            }
        }
    }

    // ---- store D: lane col = n0+j*16+l16; VGPR v -> row m0+i*16+v+8*half ----
#pragma unroll
    for (int j = 0; j < 4; ++j) {
        const int col = n0 + j * 16 + l16;
        if (col >= Nc) continue;
        const float bv = bias ? bias[col] : 0.0f;
#pragma unroll
        for (int i = 0; i < 2; ++i) {
#pragma unroll
            for (int v = 0; v < 8; ++v) {
                const int row = m0 + i * 16 + v + (half << 3);
                if (row < M) {
                    float val = acc[i][j][v] + bv;
                    if (do_relu) val = fmaxf(val, 0.0f);
                    C[(size_t)row * Nc + col] = val;
                }
            }
        }
    }
}

// ---------------------------------------------------------------------------
// Edge helpers
// ---------------------------------------------------------------------------
__global__ void zero_kernel(float* __restrict__ p, int n) {
    int i = blockIdx.x * blockDim.x + threadIdx.x;
    if (i < n) p[i] = 0.0f;
}

__global__ void edge_count_kernel(const int* __restrict__ ei, const int* __restrict__ et,
                                  float* __restrict__ cnt, int E, int n) {
    int e = blockIdx.x * blockDim.x + threadIdx.x;
    if (e >= E) return;
    int r   = et[e];
    int dst = ei[E + e];
    atomicAdd(&cnt[(size_t)r * n + dst], 1.0f);
}

__global__ void inv_count_kernel(float* __restrict__ c, int n) {
    int i = blockIdx.x * blockDim.x + threadIdx.x;
    if (i < n) c[i] = 1.0f / fmaxf(c[i], 1.0f);
}

// out[dst,f] += T[src,f] / max(deg_r[dst],1) for edges of relation `rel`.
// Vectorized: one thread handles 4 contiguous features (F % 4 == 0).
__global__ void scatter_mean_kernel(const float* __restrict__ T, const int* __restrict__ ei,
                                    const int* __restrict__ et, const float* __restrict__ invc,
                                    float* __restrict__ out, int E, int F, int n, int rel) {
    int e = blockIdx.x;
    if (et[e] != rel) return;
    int f = (blockIdx.y * blockDim.x + threadIdx.x) * 4;
    if (f >= F) return;
    int src = ei[e];
    int dst = ei[E + e];
    float w = invc[(size_t)rel * n + dst];
    v4f m = *(const v4f*)&T[(size_t)src * F + f];      // global_load_b128
    float* o = &out[(size_t)dst * F + f];
    atomicAdd(o + 0, m.x * w);
    atomicAdd(o + 1, m.y * w);
    atomicAdd(o + 2, m.z * w);
    atomicAdd(o + 3, m.w * w);
}

__global__ void relu_kernel(float* __restrict__ x, int n) {
    int i = blockIdx.x * blockDim.x + threadIdx.x;
    if (i < n) x[i] = fmaxf(x[i], 0.0f);
}

// Final: logits = h @ lin2 + b2 (K=400, 2 outputs) then log_softmax
__global__ void head_kernel(const float* __restrict__ H, const float* __restrict__ W,
                            const float* __restrict__ b, float* __restrict__ out,
                            int M, int K) {
    int i = blockIdx.x * blockDim.x + threadIdx.x;
    if (i >= M) return;
    const float* h = H + (size_t)i * K;
    float l0 = b[0], l1 = b[1];
    for (int k = 0; k < K; ++k) {
        float v = h[k];
        l0 += v * W[2 * k];
        l1 += v * W[2 * k + 1];
    }
    float m   = fmaxf(l0, l1);
    float lse = m + logf(expf(l0 - m) + expf(l1 - m));
    out[(size_t)i * 2]     = l0 - lse;
    out[(size_t)i * 2 + 1] = l1 - lse;
}

// ---------------------------------------------------------------------------
extern "C" void kernel_launch(void* const* d_in, const int* in_sizes, int n_in,
                              void* d_out, int out_size, void* d_ws, size_t ws_size,
                              hipStream_t stream) {
    const float* x     = (const float*)d_in[0];
    const int*   ei    = (const int*)  d_in[1];   // edge_index [2,E]
    const int*   et    = (const int*)  d_in[2];   // edge_type [E]
    const float* gene  = (const float*)d_in[3];
    const float* W1    = (const float*)d_in[4];   // [4,1613,1600]
    const float* root1 = (const float*)d_in[5];
    const float* b1    = (const float*)d_in[6];
    const float* W2    = (const float*)d_in[7];   // [4,1600,900]
    const float* root2 = (const float*)d_in[8];
    const float* b2    = (const float*)d_in[9];
    const float* l1w   = (const float*)d_in[10];  // [900,400]
    const float* l1b   = (const float*)d_in[11];
    const float* l2w   = (const float*)d_in[12];  // [400,2]
    const float* l2b   = (const float*)d_in[13];

    const int M = N_NODES;
    const int E = in_sizes[2];

    float* ws  = (float*)d_ws;
    size_t off = 0;
    float* T   = ws + off; off += (size_t)M * 1600;   // per-relation transform
    float* H1  = ws + off; off += (size_t)M * 1600;   // layer-1 output
    float* H2  = ws + off; off += (size_t)M * 900;    // layer-2 output
    float* H3  = ws + off; off += (size_t)M * 400;    // MLP hidden
    float* cnt = ws + off; off += (size_t)NUM_REL * M;

    const dim3 blk(256);

    // Per-(relation,dst) inverse degree
    zero_kernel<<<(NUM_REL * M + 255) / 256, blk, 0, stream>>>(cnt, NUM_REL * M);
    edge_count_kernel<<<(E + 255) / 256, blk, 0, stream>>>(ei, et, cnt, E, M);
    inv_count_kernel<<<(NUM_REL * M + 255) / 256, blk, 0, stream>>>(cnt, NUM_REL * M);

    // ---- RGCN layer 1 (1613 -> 1600) ----
    {
        dim3 g((1600 + 63) / 64, (M + 255) / 256);
        wmma_gemm_f32<<<g, blk, 0, stream>>>(x, gene, N_X, root1, b1, H1, M, 1613, 1600, 0);
        for (int r = 0; r < NUM_REL; ++r) {
            wmma_gemm_f32<<<g, blk, 0, stream>>>(x, gene, N_X,
                W1 + (size_t)r * 1613 * 1600, nullptr, T, M, 1613, 1600, 0);
            dim3 gs(E, (1600 / 4 + 255) / 256);
            scatter_mean_kernel<<<gs, blk, 0, stream>>>(T, ei, et, cnt, H1, E, 1600, M, r);
        }
        relu_kernel<<<(M * 1600 + 255) / 256, blk, 0, stream>>>(H1, M * 1600);
    }

    // ---- RGCN layer 2 (1600 -> 900) ----
    {
        dim3 g((900 + 63) / 64, (M + 255) / 256);
        wmma_gemm_f32<<<g, blk, 0, stream>>>(H1, H1, M, root2, b2, H2, M, 1600, 900, 0);
        for (int r = 0; r < NUM_REL; ++r) {
            wmma_gemm_f32<<<g, blk, 0, stream>>>(H1, H1, M,
                W2 + (size_t)r * 1600 * 900, nullptr, T, M, 1600, 900, 0);
            dim3 gs(E, (900 / 4 + 255) / 256);
            scatter_mean_kernel<<<gs, blk, 0, stream>>>(T, ei, et, cnt, H2, E, 900, M, r);
        }
        relu_kernel<<<(M * 900 + 255) / 256, blk, 0, stream>>>(H2, M * 900);
    }

    // ---- MLP head: relu(H2 @ lin1 + b) then logits + log_softmax ----
    {
        dim3 g((400 + 63) / 64, (M + 255) / 256);
        wmma_gemm_f32<<<g, blk, 0, stream>>>(H2, H2, M, l1w, l1b, H3, M, 900, 400, 1);
        head_kernel<<<(M + 255) / 256, blk, 0, stream>>>(H3, l2w, l2b, (float*)d_out, M, 400);
    }
}